// STDPConvInhibit_1675037245658
// MI455X (gfx1250) — compile-verified
//
#include <hip/hip_runtime.h>

// ---------------------------------------------------------------------------
// Locally-connected conv (unshared weights) as 3600 tiny GEMMs on CDNA5 WMMA.
//   out[b,k,i,j] = sum_{c,u,v} x[b,c,i+u,j+v] * W[i,j,c,u,v,k]
// Per pixel: M=32 (batch) x K=400 (c*25+u*5+v) x N=16 (k_out), fp32.
//
// v3: A (patches) fetched by the Tensor Data Mover as one 4-D tile
//     (v,u,c,m) = (5,5,16,32), strides (1,64,4096,65536) floats, with LDS
//     padding 2 dwords / 8 dwords -> row stride 500 (conflict-free b64
//     fragment reads). One TENSOR_LOAD_TO_LDS replaces ~3000 dynamic staging
//     instructions. B (weights) fragments load directly from global with
//     immediate offsets. Inner loop: V_WMMA_F32_16X16X4_F32.
//     (6-arg tensor_load_to_lds builtin: clang-23 / therock signature)
// ---------------------------------------------------------------------------

typedef float v2f __attribute__((ext_vector_type(2)));
typedef float v8f __attribute__((ext_vector_type(8)));
typedef unsigned int v4u __attribute__((ext_vector_type(4)));
typedef int v8i __attribute__((ext_vector_type(8)));
typedef int v4i __attribute__((ext_vector_type(4)));
typedef __attribute__((address_space(3))) int lint_t;  // LDS int*

#define NB     32     // batch
#define CIN    16
#define COUT   16
#define INHW   64
#define OUTHW  60
#define KDIM   400    // CIN*5*5, the GEMM reduction dim
#define LSTR   500    // padded LDS row stride in dwords: 400 data + 2 pad per
                      // 8 data dwords; 500 % 64 == 52 -> lane bases hit 16
                      // distinct multiples of 4 -> b64 frags use all 64 banks.

#if __has_builtin(__builtin_amdgcn_tensor_load_to_lds)
#define USE_TDM 1
#else
#define USE_TDM 0
#endif

__global__ __launch_bounds__(64)
void lc_conv_wmma_f32(const float* __restrict__ x,
                      const float* __restrict__ w,
                      float* __restrict__ out)
{
    __shared__ float Ash[NB * LSTR];    // patches A[m][pad(cuv)]

    const int j = blockIdx.x;           // output col
    const int i = blockIdx.y;           // output row
    const int t = threadIdx.x;          // 0..63 (2 waves)

#if USE_TDM
    // ---- one TDM 4-D tile load: x[m, c, i+u, j+v] -> Ash (padded) ----------
    if (t < 32) {                       // wave 0 only (wave-uniform branch)
        unsigned long long ga =
            (unsigned long long)(uintptr_t)(x + (size_t)i * INHW + j);
        unsigned lds_off =
            (unsigned)(uintptr_t)(lint_t*)(uintptr_t)&Ash[0];

        v4u g0;
        g0[0] = 1u;                                   // count=1, user mode
        g0[1] = lds_off;                              // lds_addr (bytes)
        g0[2] = (unsigned)(ga & 0xffffffffu);         // global_addr[31:0]
        g0[3] = (unsigned)((ga >> 32) & 0x1ffffffu)   // global_addr[56:32]
                | (2u << 30);                         // type = 2 ("image")

        v8i g1;
        g1[0] = (2 << 16)      // data_size = 4 bytes
              | (1 << 20)      // pad_enable
              | (2 << 22)      // pad_interval: every 8 dwords
              | (1 << 25);     // pad_amount : 2 dwords
        g1[1] = (INHW << 16);          // tensor_dim0[15:0] = 64
        g1[2] = (INHW << 16);          // tensor_dim0 hi=0 | tensor_dim1 lo=64
        g1[3] = (5 << 16);             // tensor_dim1 hi=0 | tile_dim0 = 5 (v)
        g1[4] = 5 | (CIN << 16);       // tile_dim1 = 5 (u) | tile_dim2 = 16 (c)
        g1[5] = INHW;                  // tensor_dim0_stride = 64   (u step)
        g1[6] = ((INHW * INHW) & 0xffff) << 16;  // tensor_dim1_stride lo16 (c step=4096)
        g1[7] = (INHW * INHW) >> 16;             // tensor_dim1_stride hi bits

        v4i g2;
        g2[0] = CIN;                   // tensor_dim2 = 16
        g2[1] = NB;                    // tensor_dim3 = 32
        g2[2] = CIN * INHW * INHW;     // tensor_dim2_stride = 65536 (m step)
        g2[3] = (NB << 16);            // stride hi=0 | tile_dim3 = 32 (m)

        v4i g3 = {0, 0, 0, 0};         // no dim4
        v8i g4 = {0, 0, 0, 0, 0, 0, 0, 0};  // reserved extra group (clang-23)

        __builtin_amdgcn_tensor_load_to_lds(g0, g1, g2, g3, g4, 0);
        __builtin_amdgcn_s_wait_tensorcnt(0);
    }
#else
    // fallback: plain staged copy into the same padded layout
    for (int p = t; p < NB * KDIM; p += 64) {
        int m   = p / KDIM;
        int cuv = p - m * KDIM;
        int c   = cuv / 25;
        int uv  = cuv - c * 25;
        int u   = uv / 5;
        int v   = uv - u * 5;
        Ash[m * LSTR + cuv + 2 * (cuv >> 3)] =
            x[(((size_t)m * CIN + c) * INHW + (i + u)) * INHW + (j + v)];
    }
#endif
    __syncthreads();

    // ---- WMMA loop: each wave does one 16x16 D tile over K=400 -------------
    const int wave = t >> 5;             // M-tile: batches 16*wave..16*wave+15
    const int lane = t & 31;
    const int row  = lane & 15;          // A: M row   B: N col
    const int half = lane >> 4;          // 0: K=kk,kk+1   1: K=kk+2,kk+3
    const int koff = half * 2;

    const float* ap = &Ash[(wave * 16 + row) * LSTR + koff];

    // B fragment base: w[((i*60+j)*400 + k)*16 + n]; per-lane base + imm offs.
    const float* bg = w + (size_t)(i * OUTHW + j) * (KDIM * COUT)
                        + row + half * 32;

    v8f c0 = {};
    v8f c1 = {};
    #pragma unroll 5
    for (int it = 0; it < 50; ++it) {
        const int kk = it * 8;           // data-K index
        const int kp = it * 10;          // padded LDS dword offset
        v2f a0 = *(const v2f*)(ap + kp);
        v2f b0 = { bg[kk * 16], bg[kk * 16 + 16] };
        c0 = __builtin_amdgcn_wmma_f32_16x16x4_f32(
                 false, a0, false, b0, (short)0, c0, false, false);
        v2f a1 = *(const v2f*)(ap + kp + 4);
        v2f b1 = { bg[(kk + 4) * 16], bg[(kk + 4) * 16 + 16] };
        c1 = __builtin_amdgcn_wmma_f32_16x16x4_f32(
                 false, a1, false, b1, (short)0, c1, false, false);
    }
    v8f acc = c0 + c1;

    // ---- store D: VGPR r -> M=r (lanes 0-15), M=r+8 (lanes 16-31); N=row ---
    #pragma unroll
    for (int r = 0; r < 8; ++r) {
        int b = wave * 16 + r + 8 * half;
        out[(((size_t)b * COUT + row) * OUTHW + i) * OUTHW + j] = acc[r];
    }
}

extern "C" void kernel_launch(void* const* d_in, const int* in_sizes, int n_in,
                              void* d_out, int out_size, void* d_ws, size_t ws_size,
                              hipStream_t stream) {
    (void)in_sizes; (void)n_in; (void)d_ws; (void)ws_size; (void)out_size;
    const float* x = (const float*)d_in[0];   // (32,16,64,64) fp32
    const float* w = (const float*)d_in[1];   // (60,60,16,5,5,16) fp32
    float* out = (float*)d_out;               // (32,16,60,60) fp32

    dim3 grid(OUTHW, OUTHW);
    dim3 block(64);
    lc_conv_wmma_f32<<<grid, block, 0, stream>>>(x, w, out);
}